// VIN_58823872086297
// MI455X (gfx1250) — compile-verified
//
#include <hip/hip_runtime.h>
#include <hip/hip_bf16.h>

typedef float v2f __attribute__((ext_vector_type(2)));
typedef float v8f __attribute__((ext_vector_type(8)));

#define IMSZ 128
#define PW   130            // padded (halo) row stride
#define PIX  (IMSZ * IMSZ)
#define LQ   10
#define NACT 5

// ---------------------------------------------------------------------------
// Kernel 1: collapse h-conv (150ch) + 1x1 r-conv into one effective
// 2-in / 1-out 3x3 conv: eff[0..17] = sum_c r_w[c]*h_w[c,i,dy,dx],
// eff[18] = sum_c r_w[c]*h_b[c].
// ---------------------------------------------------------------------------
__global__ void vin_effw_kernel(const float* __restrict__ h_w,
                                const float* __restrict__ h_b,
                                const float* __restrict__ r_w,
                                float* __restrict__ eff) {
    int j = threadIdx.x;
    if (j > 18) return;
    float s = 0.f;
    if (j < 18) {
        for (int c = 0; c < 150; ++c) s += r_w[c] * h_w[c * 18 + j];
    } else {
        for (int c = 0; c < 150; ++c) s += r_w[c] * h_b[c];
    }
    eff[j] = s;
}

// ---------------------------------------------------------------------------
// Kernel 2: one workgroup per image. r and double-buffered v live in LDS
// (zero halos == conv zero padding). k rounds of
//   q[ch] = conv2([r,v], wq[ch]) ; v = max_ch q
// via V_WMMA_F32_16X16X4_F32 implicit GEMM, two tiles in flight per wave.
// ---------------------------------------------------------------------------
__global__ __launch_bounds__(1024)
void vin_iter_kernel(const float* __restrict__ input,    // (B,2,128,128)
                     const int*   __restrict__ state_x,  // (B)
                     const int*   __restrict__ state_y,  // (B)
                     const int*   __restrict__ kptr,     // scalar k
                     const float* __restrict__ q_w,      // (10,1,3,3)
                     const float* __restrict__ w_w,      // (10,1,3,3)
                     const float* __restrict__ fc_w,     // (5,10)
                     const float* __restrict__ eff,      // (19)
                     float*       __restrict__ out)      // (B,5)
{
    extern __shared__ float smem[];
    float* r_s  = smem;                 // 130*130
    float* v0_s = smem + PW * PW;       // 130*130  (v, phase 0)
    float* v1_s = smem + 2 * PW * PW;   // 130*130  (v, phase 1)

    const int b    = blockIdx.x;
    const int tid  = threadIdx.x;
    const int lane = tid & 31;
    const int wave = tid >> 5;
    const int hlf  = lane >> 4;   // lane half: selects K pair within A/B frags
    const int n    = lane & 15;   // matrix row (A: channel) / col (B,D: pixel)

    // --- per-lane A-matrix fragments (weights), ISA 16x4 f32 A layout:
    //     VGPR0 holds K=4c+2*hlf, VGPR1 holds K=4c+2*hlf+1, row M = n.
    //     Columns K=18,19 and rows M>=10 are exactly zero.
    float A0[5], A1[5];
#pragma unroll
    for (int c = 0; c < 5; ++c) {
        int k0 = 4 * c + 2 * hlf;
#pragma unroll
        for (int t = 0; t < 2; ++t) {
            int k = k0 + t;
            float wv = 0.f;
            if (n < LQ && k < 18)
                wv = (k < 9) ? q_w[n * 9 + k] : w_w[n * 9 + (k - 9)];
            if (t == 0) A0[c] = wv; else A1[c] = wv;
        }
    }

    // --- zero all LDS (r halo + both v buffers; v interior starts at 0)
    for (int i = tid; i < 3 * PW * PW; i += 1024) smem[i] = 0.f;
    __syncthreads();

    // --- r = eff-conv(input), pad=1, into halo-padded LDS
    float ew[18];
#pragma unroll
    for (int i = 0; i < 18; ++i) ew[i] = eff[i];
    const float effb = eff[18];
    const float* inb = input + (size_t)b * 2 * PIX;
    for (int p = tid; p < PIX; p += 1024) {
        int y = p >> 7, x = p & 127;
        float s = effb;
#pragma unroll
        for (int i = 0; i < 2; ++i)
#pragma unroll
            for (int dy = 0; dy < 3; ++dy) {
                int yy = y + dy - 1;
                if (yy < 0 || yy >= IMSZ) continue;
#pragma unroll
                for (int dx = 0; dx < 3; ++dx) {
                    int xx = x + dx - 1;
                    if (xx < 0 || xx >= IMSZ) continue;
                    s += ew[i * 9 + dy * 3 + dx] * inb[i * PIX + yy * IMSZ + xx];
                }
            }
        r_s[(y + 1) * PW + (x + 1)] = s;
    }
    __syncthreads();

    // --- k unified VI rounds (round 0: v==0 => conv(v,w)==0 reproduces init)
    const int K = kptr[0];
    for (int it = 0; it < K; ++it) {
        const float* vin  = (it & 1) ? v1_s : v0_s;
        float*       vout = (it & 1) ? v0_s : v1_s;
#pragma unroll 1
        for (int rep = 0; rep < 16; ++rep) {
            // two 16x1 pixel tiles in flight: same row, columns 16 apart
            int t  = wave * 32 + rep * 2;
            int y  = t >> 3;
            int x0 = (t & 7) << 4;
            int base0 = y * PW + x0 + n;   // top-left of lane's 3x3 window
            int base1 = base0 + 16;

            v8f acc0 = {0.f, 0.f, 0.f, 0.f, 0.f, 0.f, 0.f, 0.f};
            v8f acc1 = {0.f, 0.f, 0.f, 0.f, 0.f, 0.f, 0.f, 0.f};
#pragma unroll
            for (int c = 0; c < 5; ++c) {
                float b00, b01, b10, b11;
#pragma unroll
                for (int t2 = 0; t2 < 2; ++t2) {
                    int k = 4 * c + 2 * hlf + t2;
                    // K>=18 has zero A-column: clamp tap, load any finite val
                    const float* sp;
                    int j;
                    if (k < 9) { sp = r_s; j = k; }
                    else       { sp = vin; j = (k - 9 > 8) ? 8 : (k - 9); }
                    int rel = (j / 3) * PW + (j % 3);
                    float va = sp[base0 + rel];
                    float vb = sp[base1 + rel];
                    if (t2 == 0) { b00 = va; b10 = vb; }
                    else         { b01 = va; b11 = vb; }
                }
                v2f aa = {A0[c], A1[c]};
                v2f bb0 = {b00, b01};
                v2f bb1 = {b10, b11};
                acc0 = __builtin_amdgcn_wmma_f32_16x16x4_f32(
                    false, aa, false, bb0, (short)0, acc0, false, false);
                acc1 = __builtin_amdgcn_wmma_f32_16x16x4_f32(
                    false, aa, false, bb1, (short)0, acc1, false, false);
            }
            // channel max, branchless: lanes 0-15 hold ch 0..7,
            // lanes 16-31 hold ch 8..15 (ch>=10 are exact zeros, excluded).
            float m8a = fmaxf(fmaxf(fmaxf(acc0[0], acc0[1]), fmaxf(acc0[2], acc0[3])),
                              fmaxf(fmaxf(acc0[4], acc0[5]), fmaxf(acc0[6], acc0[7])));
            float m8b = fmaxf(fmaxf(fmaxf(acc1[0], acc1[1]), fmaxf(acc1[2], acc1[3])),
                              fmaxf(fmaxf(acc1[4], acc1[5]), fmaxf(acc1[6], acc1[7])));
            float ma = hlf ? fmaxf(acc0[0], acc0[1]) : m8a;
            float mb = hlf ? fmaxf(acc1[0], acc1[1]) : m8b;
            ma = fmaxf(ma, __shfl_xor(ma, 16, 32));
            mb = fmaxf(mb, __shfl_xor(mb, 16, 32));
            if (hlf == 0) {
                vout[(y + 1) * PW + (x0 + n + 1)]      = ma;
                vout[(y + 1) * PW + (x0 + 16 + n + 1)] = mb;
            }
        }
        __syncthreads();
    }

    // --- final q only at (state_x, state_y), then FC -> logits
    const float* vf = (K & 1) ? v1_s : v0_s;
    if (tid < NACT) {
        int sx = state_x[b], sy = state_y[b];   // sx = row (H), sy = col (W)
        int base = sx * PW + sy;                // top-left of 3x3 in halo'd map
        float logit = 0.f;
        for (int ch = 0; ch < LQ; ++ch) {
            float q = 0.f;
#pragma unroll
            for (int j = 0; j < 9; ++j) {
                int dy = j / 3, dx = j % 3;
                float rr = r_s[base + dy * PW + dx];
                float vv = vf [base + dy * PW + dx];
                q += q_w[ch * 9 + j] * rr + w_w[ch * 9 + j] * vv;
            }
            logit += fc_w[tid * LQ + ch] * q;
        }
        out[b * NACT + tid] = logit;
    }
}

extern "C" void kernel_launch(void* const* d_in, const int* in_sizes, int n_in,
                              void* d_out, int out_size, void* d_ws, size_t ws_size,
                              hipStream_t stream) {
    const float* input = (const float*)d_in[0];
    const int*   sx    = (const int*)  d_in[1];
    const int*   sy    = (const int*)  d_in[2];
    const int*   kk    = (const int*)  d_in[3];
    const float* h_w   = (const float*)d_in[4];
    const float* h_b   = (const float*)d_in[5];
    const float* r_w   = (const float*)d_in[6];
    const float* q_w   = (const float*)d_in[7];
    const float* w_w   = (const float*)d_in[8];
    const float* fc_w  = (const float*)d_in[9];

    float* eff = (float*)d_ws;                    // 19 floats of scratch
    const int B = in_sizes[1];                    // 64 images

    vin_effw_kernel<<<1, 32, 0, stream>>>(h_w, h_b, r_w, eff);

    size_t shbytes = (size_t)3 * PW * PW * sizeof(float);   // 202800 B < 320 KB
    vin_iter_kernel<<<B, 1024, shbytes, stream>>>(
        input, sx, sy, kk, q_w, w_w, fc_w, eff, (float*)d_out);
}